// RandomEQ_34531537060272
// MI455X (gfx1250) — compile-verified
//
#include <hip/hip_runtime.h>

typedef float v2f __attribute__((ext_vector_type(2)));
typedef float v8f __attribute__((ext_vector_type(8)));

#define SR      32000.0f
#define T_LEN   160000
#define TILE    256               // samples per tile (16 blocks of 16)
#define NTILES  (T_LEN / TILE)    // 625 exactly

// padded LDS index: linear sample i -> i + i/16  (stride 17 per 16-block,
// makes the stride-16 column reads bank-conflict free on 64 banks)
__device__ __forceinline__ int pidx(int i) { return i + (i >> 4); }

// wave-uniform broadcast from a fixed lane: v_readlane_b32 (no DS round trip)
__device__ __forceinline__ float rdlane(float v, int l) {
  return __int_as_float(__builtin_amdgcn_readlane(__float_as_int(v), l));
}

__launch_bounds__(32)
__global__ void eq3_wmma_kernel(const float* __restrict__ clip,
                                const float* __restrict__ freqs,
                                const float* __restrict__ gains,
                                const float* __restrict__ qs,
                                float* __restrict__ out) {
  __shared__ float xb[TILE + 16];           // padded tile buffer (272 floats)

  const int  lane = threadIdx.x;            // 0..31 (wave32)
  const bool hi   = lane >= 16;
  const int  jcol = lane & 15;              // A-matrix M row / B,D-matrix N col
  const long rbase = (long)blockIdx.x * T_LEN;

  // ---------------- per-stage filter setup (all lanes, once) ----------------
  float b0s[3], b1s[3], b2s[3];
  v2f   afrag[3][4];                        // A fragments for 4 chained K=4 WMMAs
  float gl[3][8], kl[3][8];                 // per-lane homogeneous responses g[M],k[M]
  float g14s[3], g15s[3], k14s[3], k15s[3]; // wave-uniform scan coefficients (M)
  float m11[3], m12[3], m21[3], m22[3];     // M^2 for the step-doubled scan

  #pragma unroll
  for (int st = 0; st < 3; ++st) {
    const float f = freqs[st], gdb = gains[st], q = qs[st];
    const float w0 = 2.0f * 3.14159265358979323846f * f / SR;
    const float A  = expf(gdb * (0.025f * logf(10.0f)));   // 10^(g/40)
    const float sw = sinf(w0), cw = cosf(w0);
    const float alpha = sw / (2.0f * q);
    float b0 = 1.0f + alpha * A;
    float b1 = -2.0f * cw;
    float b2 = 1.0f - alpha * A;
    const float a0  = 1.0f + alpha / A;
    float a1 = -2.0f * cw;
    float a2 = 1.0f - alpha / A;
    const float ra0 = 1.0f / a0;
    b0 *= ra0; b1 *= ra0; b2 *= ra0; a1 *= ra0; a2 *= ra0;
    b0s[st] = b0; b1s[st] = b1; b2s[st] = b2;

    #pragma unroll
    for (int t = 0; t < 4; ++t) { afrag[st][t][0] = 0.f; afrag[st][t][1] = 0.f; }

    // h: AR impulse response (h[0]=1); g: response to y[-1]=1; k: to y[-2]=1
    float h1 = 0.f, h2 = 0.f;
    float G1 = 1.f, G2 = 0.f;
    float K1 = 0.f, K2 = 1.f;
    #pragma unroll
    for (int i = 0; i < 16; ++i) {
      const float hc = (i == 0) ? 1.0f : (-a1 * h1 - a2 * h2);
      const float gc = -a1 * G1 - a2 * G2;
      const float kc = -a1 * K1 - a2 * K2;
      // A-fragment grab: afrag[t][c] = T[M][k] = h[M-k], M=jcol,
      // k = 4t + (hi?2:0) + c   (A layout: lane half picks K pair, VGPR picks K)
      #pragma unroll
      for (int t = 0; t < 4; ++t) {
        const int kb = 4 * t + (hi ? 2 : 0);
        if (jcol - kb       == i) afrag[st][t][0] = hc;
        if (jcol - (kb + 1) == i) afrag[st][t][1] = hc;
      }
      // correction grab: gl[r] = g[r + hi*8]  (D layout: VGPR r holds M=r(+8))
      #pragma unroll
      for (int r = 0; r < 8; ++r) {
        if (i == r + (hi ? 8 : 0)) { gl[st][r] = gc; kl[st][r] = kc; }
      }
      if (i == 14) { g14s[st] = gc; k14s[st] = kc; }
      if (i == 15) { g15s[st] = gc; k15s[st] = kc; }
      h2 = h1; h1 = hc;  G2 = G1; G1 = gc;  K2 = K1; K1 = kc;
    }
    // M = [[g15,k15],[g14,k14]];  M2 = M*M (wave-uniform per stage)
    m11[st] = g15s[st] * g15s[st] + k15s[st] * g14s[st];
    m12[st] = g15s[st] * k15s[st] + k15s[st] * k14s[st];
    m21[st] = g14s[st] * g15s[st] + k14s[st] * g14s[st];
    m22[st] = g14s[st] * k15s[st] + k14s[st] * k14s[st];
  }

  // recurrence state (y[-1], y[-2]) per stage; raw-input FIR history for stage 0
  float s1v[3] = {0.f, 0.f, 0.f}, s2v[3] = {0.f, 0.f, 0.f};
  float xh1 = 0.f, xh2 = 0.f;

  for (int tile = 0; tile < NTILES; ++tile) {
    const long tbase = rbase + (long)tile * TILE;

    // coalesced b128 load of 256 samples -> padded LDS
    const float4 xa = ((const float4*)(clip + tbase))[2 * lane + 0];
    const float4 xc = ((const float4*)(clip + tbase))[2 * lane + 1];
    xb[pidx(8 * lane + 0)] = xa.x;  xb[pidx(8 * lane + 1)] = xa.y;
    xb[pidx(8 * lane + 2)] = xa.z;  xb[pidx(8 * lane + 3)] = xa.w;
    xb[pidx(8 * lane + 4)] = xc.x;  xb[pidx(8 * lane + 5)] = xc.y;
    xb[pidx(8 * lane + 6)] = xc.z;  xb[pidx(8 * lane + 7)] = xc.w;

    if (tile + 1 < NTILES)                           // hide next tile's latency
      __builtin_prefetch(clip + tbase + TILE, 0, 0);

    // pre-tile FIR histories: stage k>0 consumes stage k-1's pre-tile state
    float ih1[3], ih2[3];
    ih1[0] = xh1;    ih2[0] = xh2;
    ih1[1] = s1v[0]; ih2[1] = s2v[0];
    ih1[2] = s1v[1]; ih2[2] = s2v[1];
    // raw history for the next tile (intra-wave DS ordering covers the RAW)
    xh1 = xb[pidx(255)];  xh2 = xb[pidx(254)];

    #pragma unroll
    for (int st = 0; st < 3; ++st) {
      const float b0 = b0s[st], b1 = b1s[st], b2 = b2s[st];
      const float H1 = ih1[st], H2 = ih2[st];

      // B fragments: U[s][j] = FIR of stage input at tile sample 16j+s
      v2f bfrag[4];
      #pragma unroll
      for (int t = 0; t < 4; ++t) {
        const int kb   = 4 * t + (hi ? 2 : 0);
        const int base = 16 * jcol + kb - 2;         // < 0 only for lane 0, t 0
        const float v0 = (base + 0 < 0) ? H2 : xb[pidx(base + 0)];
        const float v1 = (base + 1 < 0) ? H1 : xb[pidx(base + 1)];
        const float v2 = xb[pidx(base + 2)];
        const float v3 = xb[pidx(base + 3)];
        bfrag[t][0] = b0 * v2 + b1 * v1 + b2 * v0;
        bfrag[t][1] = b0 * v3 + b1 * v2 + b2 * v1;
      }

      // zero-state block responses: Y0 = T(16x16) * U(16x16) via 4 K=4 WMMAs
      v8f acc = {};
      #pragma unroll
      for (int t = 0; t < 4; ++t) {
        acc = __builtin_amdgcn_wmma_f32_16x16x4_f32(
            false, afrag[st][t], false, bfrag[t], (short)0, acc, false, false);
      }

      // Broadcast rows M=14,15 of Y0 (lanes 16..31 hold them in VGPRs 6,7)
      // to all lanes via v_readlane -- pure VALU, no DS round trips.
      float y15a[16], y14a[16];
      #pragma unroll
      for (int j = 0; j < 16; ++j) {
        y15a[j] = rdlane(acc[7], 16 + j);
        y14a[j] = rdlane(acc[6], 16 + j);
      }

      const float g14 = g14s[st], g15 = g15s[st];
      const float k14 = k14s[st], k15 = k15s[st];

      // Step-doubled affine scan: S_{j+2} = M^2 S_j + (M y_j + y_{j+1}).
      // Drive terms d_m are independent FMAs (no serial dependency).
      float d1[8], d2[8];
      #pragma unroll
      for (int m = 0; m < 8; ++m) {
        const float p15 = y15a[2 * m], p14 = y14a[2 * m];
        d1[m] = g15 * p15 + k15 * p14 + y15a[2 * m + 1];
        d2[m] = g14 * p15 + k14 * p14 + y14a[2 * m + 1];
      }
      const float A11 = m11[st], A12 = m12[st];
      const float A21 = m21[st], A22 = m22[st];

      float S1 = s1v[st], S2 = s2v[st];
      float Se1 = 0.f, Se2 = 0.f, cy15 = 0.f, cy14 = 0.f;
      const int jh = jcol >> 1;
      #pragma unroll
      for (int m = 0; m < 8; ++m) {
        if (m == jh) {                       // incoming state of my even block
          Se1 = S1; Se2 = S2;
          cy15 = y15a[2 * m]; cy14 = y14a[2 * m];
        }
        const float nS1 = d1[m] + A11 * S1 + A12 * S2;
        const float nS2 = d2[m] + A21 * S1 + A22 * S2;
        S1 = nS1; S2 = nS2;
      }
      s1v[st] = S1; s2v[st] = S2;            // y[255], y[254] of this tile

      // odd block: advance the captured even-block state one step (parallel)
      const bool odd = (jcol & 1) != 0;
      const float o1 = cy15 + g15 * Se1 + k15 * Se2;
      const float o2 = cy14 + g14 * Se1 + k14 * Se2;
      const float mys1 = odd ? o1 : Se1;
      const float mys2 = odd ? o2 : Se2;

      // homogeneous correction, then stage output -> LDS (reads all done)
      #pragma unroll
      for (int r = 0; r < 8; ++r)
        acc[r] += gl[st][r] * mys1 + kl[st][r] * mys2;
      #pragma unroll
      for (int r = 0; r < 8; ++r)
        xb[pidx(16 * jcol + r + (hi ? 8 : 0))] = acc[r];
    }

    // coalesced b128 store of the finished tile
    float4 o0, o1;
    o0.x = xb[pidx(8 * lane + 0)];  o0.y = xb[pidx(8 * lane + 1)];
    o0.z = xb[pidx(8 * lane + 2)];  o0.w = xb[pidx(8 * lane + 3)];
    o1.x = xb[pidx(8 * lane + 4)];  o1.y = xb[pidx(8 * lane + 5)];
    o1.z = xb[pidx(8 * lane + 6)];  o1.w = xb[pidx(8 * lane + 7)];
    ((float4*)(out + tbase))[2 * lane + 0] = o0;
    ((float4*)(out + tbase))[2 * lane + 1] = o1;
  }
}

extern "C" void kernel_launch(void* const* d_in, const int* in_sizes, int n_in,
                              void* d_out, int out_size, void* d_ws, size_t ws_size,
                              hipStream_t stream) {
  const float* clip  = (const float*)d_in[0];   // [128, 160000] f32
  const float* freqs = (const float*)d_in[1];   // [3]
  const float* gains = (const float*)d_in[2];   // [3]
  const float* qs    = (const float*)d_in[3];   // [3]
  float* out = (float*)d_out;
  const int rows = in_sizes[0] / T_LEN;         // 128
  eq3_wmma_kernel<<<dim3(rows), dim3(32), 0, stream>>>(clip, freqs, gains, qs, out);
}